// series_decomp_57346403336707
// MI455X (gfx1250) — compile-verified
//
#include <hip/hip_runtime.h>
#include <hip/hip_bf16.h>

// ---------------------------------------------------------------------------
// series_decomp on MI455X (gfx1250)
//
// x: [B=32, L=2048, N=512] f32.  Rows = (b, n) pairs along t (stride N).
//   forward rFFT  -> WMMA f32 GEMM  [16384 x 2048] x [2048 x 2x1025]
//                    (V_WMMA_F32_16X16X4_F32, fp32-exact; A tiles staged into
//                     LDS by the Tensor Data Mover, double-buffered)
//   trend         -> bins 0..24
//   seasonal      -> per-row top-64 of bins 25..1024 (compacted, VALU inverse)
// Twiddles: exact LDS table cos(2*pi*i/2048); -sin(th) = cos(th + pi/2)
//           (index +512 mod 2048), phase index = (f*t) & 2047.
//
// Workspace layout (floats), ~149 MB (fits device; spectrum is L2-resident):
//   costab [2048] | spec_re [16384 x 1040] | spec_im [16384 x 1040]
//   | tk_re [16384 x 64] | tk_im [16384 x 64] | tk_f(int) [16384 x 64]
// ---------------------------------------------------------------------------

typedef __attribute__((ext_vector_type(2))) float v2f;
typedef __attribute__((ext_vector_type(8))) float v8f;
typedef __attribute__((ext_vector_type(4))) unsigned int u32x4;
typedef __attribute__((ext_vector_type(8))) int i32x8;
typedef __attribute__((ext_vector_type(4))) int i32x4;

#define NB    32
#define LEN   2048
#define NCH   512
#define ROWS  (NB * NCH)        // 16384
#define NF    1025
#define F_PAD 1040              // 65 tiles of 16
#define LOWF  25
#define KTOP  64
#define CHUNK_T 256
#define NCHUNK  (LEN / CHUNK_T) // 8
#define WAVES_PER_BLOCK 8

// ---------------------------------------------------------------- kernel 0 --
__global__ void build_costab_k(float* __restrict__ costab) {
    int i = blockIdx.x * blockDim.x + threadIdx.x;
    if (i < LEN) costab[i] = cospif((float)i * (1.0f / 1024.0f)); // cos(2*pi*i/2048)
}

// --------------------------------------------------------------- TDM issue --
// One descriptor = one 2D tile: tile_dim0 = 16 floats (contiguous channels),
// tile_dim1 = 256 rows (t), row stride = 512 floats (= NCH). The TDM packs the
// tile linearly into LDS as xs[t_local*16 + n_local] (X fastest, then Y).
// D# bitfields per CDNA5 ISA.
__device__ __forceinline__ void tdm_load_tile(unsigned long long gaddr,
                                              unsigned int lds_addr) {
    u32x4 g0 = {
        1u,                                        // [1:0] count=1 (valid D#)
        lds_addr,                                  // [63:32] lds_addr (bytes)
        (unsigned int)gaddr,                       // [95:64] global_addr lo
        ((unsigned int)(gaddr >> 32) & 0x01FFFFFFu)// [120:96] global_addr hi
            | (2u << 30)                           // [127:126] type=2 (image)
    };
    i32x8 g1 = {
        (int)(2u << 16),    // [17:16] data_size=2 -> 4 bytes; mask/flags = 0
        (int)(16u << 16),   // [63:48]  tensor_dim0 lo16 = 16
        (int)(256u << 16),  // [79:64]  tensor_dim0 hi = 0 | [95:80] tensor_dim1 lo16 = 256
        (int)(16u << 16),   // [111:96] tensor_dim1 hi = 0 | [127:112] tile_dim0 = 16
        256,                // [143:128] tile_dim1 = 256 | [159:144] tile_dim2 = 0
        512,                // [191:160] tensor_dim0_stride lo32 = 512 elements
        0,                  // [207:192] stride hi = 0 | dim1_stride lo16 = 0
        0                   // dim1_stride hi = 0 (unused, 2D tile)
    };
    i32x4 gz = {};          // groups 2/3: zero (tensor <= 2D)
#if defined(__clang_major__) && __clang_major__ >= 23
    i32x8 g4 = {};
    __builtin_amdgcn_tensor_load_to_lds(g0, g1, gz, gz, g4, 0);
#else
    __builtin_amdgcn_tensor_load_to_lds(g0, g1, gz, gz, 0);
#endif
}

// ---------------------------------------------------------------- kernel 1 --
// Forward DFT via V_WMMA_F32_16X16X4_F32, x tiles DMA'd by the TDM.
// blockIdx.x = row tile (16 consecutive channels of one batch)
// blockIdx.y = group of 8 freq tiles (one per wave)
//
// Assumed f32 WMMA operand layouts (ISA 7.12.2):
//   A 16x4 : v0 = {lanes0-15: K=0, lanes16-31: K=2}, v1 = {K=1, K=3}, M = lane&15
//   B 4x16 : mirrored (v0 = {K=0 | K=2}, v1 = {K=1 | K=3}), N = lane&15
//   D      : VGPR r -> M = r + (lane<16 ? 0 : 8), N = lane&15
__global__ void fwd_dft_wmma_k(const float* __restrict__ x,
                               const float* __restrict__ costab_g,
                               float* __restrict__ spec_re,
                               float* __restrict__ spec_im) {
    extern __shared__ float lds[];
    float* costab = lds;                       // [2048]
    float* xs0    = lds + LEN;                 // [CHUNK_T][16] buffer 0
    float* xs1    = xs0 + CHUNK_T * 16;        // [CHUNK_T][16] buffer 1

    const int tid  = threadIdx.x;
    const int wave = tid >> 5;
    const int lane = tid & 31;

    const int row0  = blockIdx.x * 16;         // global row = b*NCH + n
    const int b     = row0 >> 9;               // /512
    const int n0    = row0 & (NCH - 1);
    const int ftile = blockIdx.y * WAVES_PER_BLOCK + wave;
    const int fbase = ftile * 16;
    const bool active = (fbase < NF);

    for (int i = tid; i < LEN; i += 256) costab[i] = costab_g[i];

    const int m   = lane & 15;                 // A: row-in-tile; B: freq col
    const int khi = (lane >> 4) * 2;           // this lane's K sub-pair offset
    const int f   = fbase + m;                 // frequency for B operand lanes

    v8f c_re = {};
    v8f c_im = {};

    // Flat->LDS aperture: low 32 bits of a shared pointer are the LDS offset.
    const unsigned int lds_xs0 = (unsigned int)(unsigned long long)(uintptr_t)xs0;
    const unsigned int lds_xs1 = (unsigned int)(unsigned long long)(uintptr_t)xs1;
    const unsigned long long gbase =
        (unsigned long long)(uintptr_t)(x + (size_t)b * LEN * NCH + n0);
    const unsigned long long chunk_bytes = (unsigned long long)CHUNK_T * NCH * 4;

    if (tid < 32) tdm_load_tile(gbase, lds_xs0);   // prefetch chunk 0

    for (int chunk = 0; chunk < NCHUNK; ++chunk) {
        if (tid < 32) {
            if (chunk + 1 < NCHUNK) {
                tdm_load_tile(gbase + (unsigned long long)(chunk + 1) * chunk_bytes,
                              ((chunk + 1) & 1) ? lds_xs1 : lds_xs0);
                __builtin_amdgcn_s_wait_tensorcnt(1);  // in-order: chunk c resident
            } else {
                __builtin_amdgcn_s_wait_tensorcnt(0);
            }
        }
        __syncthreads();                      // tile visible to all 8 waves

        const float* xs = (chunk & 1) ? xs1 : xs0;
        if (active) {
            for (int k0 = 0; k0 < CHUNK_T; k0 += 4) {
                const int ta = k0 + khi;               // local t of lane's K pair
                v2f a, br, bi;
                a.x = xs[ta * 16 + m];                 // banks 0-15/32-47: clean
                a.y = xs[(ta + 1) * 16 + m];

                const int tg = chunk * CHUNK_T + ta;   // global t
                const int p0 = (f * tg) & (LEN - 1);
                const int p1 = (p0 + f) & (LEN - 1);
                br.x = costab[p0];                     //  cos
                br.y = costab[p1];
                bi.x = costab[(p0 + 512) & (LEN - 1)]; // -sin
                bi.y = costab[(p1 + 512) & (LEN - 1)];

                c_re = __builtin_amdgcn_wmma_f32_16x16x4_f32(
                    false, a, false, br, (short)0, c_re, false, false);
                c_im = __builtin_amdgcn_wmma_f32_16x16x4_f32(
                    false, a, false, bi, (short)0, c_im, false, false);
            }
        }
        __syncthreads();                      // protect buffer before next DMA
    }

    if (active) {
        const int mrow = (lane < 16) ? 0 : 8;
        const int fcol = fbase + (lane & 15);
        if (fcol < NF) {
#pragma unroll
            for (int r = 0; r < 8; ++r) {
                size_t off = (size_t)(row0 + r + mrow) * F_PAD + fcol;
                spec_re[off] = c_re[r];
                spec_im[off] = c_im[r];
            }
        }
    }
}

// ---------------------------------------------------------------- kernel 2 --
// Per-row top-64 of |F|^2 over bins 25..1024; writes compacted, pre-scaled
// seasonal coefficients (x 2/L, Nyquist x 1/L).
__global__ void topk_select_k(const float* __restrict__ spec_re,
                              const float* __restrict__ spec_im,
                              float* __restrict__ tk_re,
                              float* __restrict__ tk_im,
                              int*   __restrict__ tk_f) {
    __shared__ float mag[1024];
    __shared__ float rval[256];
    __shared__ int   ridx[256];

    const int row = blockIdx.x;
    const int tid = threadIdx.x;
    const float* re = spec_re + (size_t)row * F_PAD;
    const float* im = spec_im + (size_t)row * F_PAD;

    for (int i = tid; i < 1024; i += 256) {
        int fg = LOWF + i;                       // 25..1048
        float v = -1.0f;                          // pad below any real |F|^2
        if (fg < NF) { float a = re[fg], c = im[fg]; v = a * a + c * c; }
        mag[i] = v;
    }
    __syncthreads();

    for (int sel = 0; sel < KTOP; ++sel) {
        float bv = -2.0f; int bi_ = 0;
        for (int i = tid; i < 1024; i += 256) {
            float v = mag[i];
            if (v > bv) { bv = v; bi_ = i; }
        }
        rval[tid] = bv; ridx[tid] = bi_;
        __syncthreads();
        for (int s = 128; s > 0; s >>= 1) {
            if (tid < s && rval[tid + s] > rval[tid]) {
                rval[tid] = rval[tid + s]; ridx[tid] = ridx[tid + s];
            }
            __syncthreads();
        }
        if (tid == 0) {
            int w  = ridx[0];
            int fg = LOWF + w;
            float coef = ((fg == LEN / 2) ? 1.0f : 2.0f) * (1.0f / (float)LEN);
            tk_re[(size_t)row * KTOP + sel] = re[fg] * coef;
            tk_im[(size_t)row * KTOP + sel] = im[fg] * coef;
            tk_f [(size_t)row * KTOP + sel] = fg;
            mag[w] = -3.0f;                      // remove winner
        }
        __syncthreads();
    }
}

// ---------------------------------------------------------------- kernel 3 --
// Inverse synthesis: trend (25 bins) + seasonal (64 gathered bins).
// x[t] = sum coef_f * (Re_f*cos(2*pi*f*t/L) - Im_f*sin(...)),
// coef = 1/L for f==0 and f==L/2, else 2/L.  -sin -> costab[(p+512)&2047].
// Block: 16 rows (consecutive channels) x 16 t; stores 64B-coalesced over n.
__global__ void inverse_k(const float* __restrict__ spec_re,
                          const float* __restrict__ spec_im,
                          const float* __restrict__ tk_re,
                          const float* __restrict__ tk_im,
                          const int*   __restrict__ tk_f,
                          const float* __restrict__ costab_g,
                          float* __restrict__ out) {
    extern __shared__ float lds[];
    float* costab = lds;                         // [2048]
    float* lre = costab + LEN;                   // [16][25]
    float* lim = lre + 16 * LOWF;                // [16][25]
    float* sre = lim + 16 * LOWF;                // [16][64]
    float* sim = sre + 16 * KTOP;                // [16][64]
    int*   sfr = (int*)(sim + 16 * KTOP);        // [16][64]

    const int tid  = threadIdx.x;
    const int row0 = blockIdx.x * 16;
    const int b    = row0 >> 9;
    const int n0   = row0 & (NCH - 1);

    for (int i = tid; i < LEN; i += 256) costab[i] = costab_g[i];
    for (int i = tid; i < 16 * LOWF; i += 256) {
        int rl = i / LOWF, fg = i % LOWF;
        float coef = ((fg == 0) ? 1.0f : 2.0f) * (1.0f / (float)LEN);
        lre[i] = spec_re[(size_t)(row0 + rl) * F_PAD + fg] * coef;
        lim[i] = spec_im[(size_t)(row0 + rl) * F_PAD + fg] * coef;
    }
    for (int i = tid; i < 16 * KTOP; i += 256) {
        int rl = i >> 6, k = i & 63;
        size_t src = (size_t)(row0 + rl) * KTOP + k;
        sre[i] = tk_re[src];
        sim[i] = tk_im[src];
        sfr[i] = tk_f[src];
    }
    __syncthreads();

    const int nl = tid & 15;                     // channel within tile
    const int tl = tid >> 4;                     // t within 16-t tile
    float* trend = out;
    float* seas  = out + (size_t)NB * LEN * NCH;
    const size_t obase = (size_t)b * LEN * NCH + n0 + nl;

    for (int tt = 0; tt < LEN; tt += 16) {
        const int t = tt + tl;
        float accT = 0.0f, accS = 0.0f;
#pragma unroll 5
        for (int fg = 0; fg < LOWF; ++fg) {
            int p = (fg * t) & (LEN - 1);
            accT += lre[nl * LOWF + fg] * costab[p]
                  + lim[nl * LOWF + fg] * costab[(p + 512) & (LEN - 1)];
        }
#pragma unroll 4
        for (int k = 0; k < KTOP; ++k) {
            int fg = sfr[nl * KTOP + k];
            int p  = (fg * t) & (LEN - 1);
            accS += sre[nl * KTOP + k] * costab[p]
                  + sim[nl * KTOP + k] * costab[(p + 512) & (LEN - 1)];
        }
        trend[obase + (size_t)t * NCH] = accT;
        seas [obase + (size_t)t * NCH] = accS;
    }
}

// ------------------------------------------------------------------ launch --
extern "C" void kernel_launch(void* const* d_in, const int* in_sizes, int n_in,
                              void* d_out, int out_size, void* d_ws, size_t ws_size,
                              hipStream_t stream) {
    const float* x = (const float*)d_in[0];
    float* out = (float*)d_out;

    float* ws      = (float*)d_ws;
    float* costab  = ws;
    float* spec_re = costab + LEN;
    float* spec_im = spec_re + (size_t)ROWS * F_PAD;
    float* tk_re   = spec_im + (size_t)ROWS * F_PAD;
    float* tk_im   = tk_re + (size_t)ROWS * KTOP;
    int*   tk_f    = (int*)(tk_im + (size_t)ROWS * KTOP);

    build_costab_k<<<(LEN + 255) / 256, 256, 0, stream>>>(costab);

    dim3 g1(ROWS / 16, (NF / 16 + WAVES_PER_BLOCK) / WAVES_PER_BLOCK); // 1024 x 9
    size_t lds1 = (size_t)(LEN + 2 * CHUNK_T * 16) * sizeof(float);    // 8KB + 2x16KB
    fwd_dft_wmma_k<<<g1, 256, lds1, stream>>>(x, costab, spec_re, spec_im);

    topk_select_k<<<ROWS, 256, 0, stream>>>(spec_re, spec_im, tk_re, tk_im, tk_f);

    size_t lds3 = (size_t)(LEN + 2 * 16 * LOWF + 3 * 16 * KTOP) * sizeof(float);
    inverse_k<<<ROWS / 16, 256, lds3, stream>>>(spec_re, spec_im,
                                                tk_re, tk_im, tk_f, costab, out);
}